// BinaryVectorQuantizer_78924319031719
// MI455X (gfx1250) — compile-verified
//
#include <hip/hip_runtime.h>

typedef __attribute__((ext_vector_type(2))) float v2f;
typedef __attribute__((ext_vector_type(8))) float v8f;

static constexpr int kL       = 12;
static constexpr int kRows    = 65536;
static constexpr int kQElems  = kRows * kL;     // 786432 floats of quantized output
static constexpr int kLossOff = kQElems;        // scalar commitment loss
static constexpr int kIdxOff  = kQElems + 1;    // 65536 encoding indices (as float)

// Sum (chunks*64) consecutive floats with chained V_WMMA_F32_16X16X4_F32.
// Must run on a full wave32 (EXEC all ones). Every lane returns the total.
// A-tile layout (16x4 f32): lane L<16 holds A[L][0..1], lane L>=16 holds A[L-16][2..3].
// With B == ones(4x16), D[m][n] = sum_k A[m][k] for every n, accumulated into C.
__device__ __forceinline__ float wave_wmma_sum(const float* __restrict__ vals,
                                               int chunks, int lane) {
  const int m     = lane & 15;
  const int kbase = (lane < 16) ? 0 : 2;
  v2f ones; ones[0] = 1.0f; ones[1] = 1.0f;
  v8f acc = {0.f, 0.f, 0.f, 0.f, 0.f, 0.f, 0.f, 0.f};
  for (int c = 0; c < chunks; ++c) {
    const float* p = vals + c * 64 + m * 4 + kbase;
    v2f a; a[0] = p[0]; a[1] = p[1];
    acc = __builtin_amdgcn_wmma_f32_16x16x4_f32(false, a, false, ones,
                                                (short)0, acc, false, false);
  }
  // Lanes 0-15 hold D[0..7][n] in acc[0..7]; lanes 16-31 hold D[8..15][n].
  float t = 0.f;
#pragma unroll
  for (int i = 0; i < 8; ++i) t += acc[i];
  t += __shfl_xor(t, 16, 32);   // combine the two row halves
  return t;
}

// One lane per row: threshold each dim at 0.5 (exact argmin over the binary
// codebook; strict '>' matches argmin's lowest-index tie break), pack the
// index MSB-first, write quantized bits, and reduce per-row squared error.
__global__ __launch_bounds__(256)
void vq_main(const float* __restrict__ x, float* __restrict__ out,
             float* __restrict__ ws) {
  __shared__ float errs[256];
  const int tid = threadIdx.x;
  const int row = blockIdx.x * 256 + tid;

  const float4* xr = reinterpret_cast<const float4*>(x + (size_t)row * kL);
  float4 a = xr[0], b = xr[1], c = xr[2];
  float v[kL] = {a.x, a.y, a.z, a.w, b.x, b.y, b.z, b.w, c.x, c.y, c.z, c.w};

  unsigned idx = 0u;
  float err = 0.f;
  float q[kL];
#pragma unroll
  for (int j = 0; j < kL; ++j) {
    const bool bit = v[j] > 0.5f;          // x==0.5 -> bit 0 (lowest-index tie)
    const float qj = bit ? 1.0f : 0.0f;
    const float d  = qj - v[j];
    err += d * d;
    idx = (idx << 1) | (unsigned)bit;      // j=0 is the MSB
    q[j] = qj;
  }

  float4* qo = reinterpret_cast<float4*>(out + (size_t)row * kL);
  qo[0] = make_float4(q[0], q[1], q[2],  q[3]);
  qo[1] = make_float4(q[4], q[5], q[6],  q[7]);
  qo[2] = make_float4(q[8], q[9], q[10], q[11]);
  out[kIdxOff + row] = (float)idx;

  errs[tid] = err;
  __syncthreads();
  if (tid < 32) {                          // wave 0, all 32 lanes active
    const float total = wave_wmma_sum(errs, 4, tid);   // 256 values
    if (tid == 0) ws[blockIdx.x] = total;
  }
}

// Single-wave deterministic reduction of the 256 block partials -> mean.
__global__ __launch_bounds__(32)
void vq_reduce(const float* __restrict__ ws, float* __restrict__ out) {
  const int tid = threadIdx.x;
  const float total = wave_wmma_sum(ws, 4, tid);       // 256 partials
  if (tid == 0) out[kLossOff] = total * (1.0f / (float)kQElems);
}

extern "C" void kernel_launch(void* const* d_in, const int* in_sizes, int n_in,
                              void* d_out, int out_size, void* d_ws, size_t ws_size,
                              hipStream_t stream) {
  (void)in_sizes; (void)n_in; (void)out_size; (void)ws_size;
  const float* x  = (const float*)d_in[0];   // [65536, 12] inputs
  // d_in[1] (codebook) is unused: its structure (all 12-bit codes) is exploited
  // analytically, which is what makes this kernel memory-bound instead of a
  // 6.4 GFLOP GEMM + 4096-way argmin.
  float* out = (float*)d_out;
  float* ws  = (float*)d_ws;                 // 256 block partial sums

  vq_main  <<<kRows / 256, 256, 0, stream>>>(x, out, ws);
  vq_reduce<<<1, 32, 0, stream>>>(ws, out);
}